// SelfAttention_24670292148260
// MI455X (gfx1250) — compile-verified
//
#include <hip/hip_runtime.h>
#include <hip/hip_bf16.h>
#include <math.h>

// ---------------------------------------------------------------------------
// Problem constants (fp32 self-attention encoder layer)
// ---------------------------------------------------------------------------
#define BATCH     8
#define S_LEN     1025            // grid 32x32 + 1 global node
#define S_TILES   65              // ceil(1025/16)
#define S_PAD     (S_TILES * 16)  // 1040
#define HIDDEN    512
#define N_HEAD    8
#define HEAD_DIM  64
#define M_ROWS    (BATCH * S_LEN) // 8200
#define PROJ_N    (3 * HIDDEN)    // 1536

typedef float v2f __attribute__((ext_vector_type(2)));
typedef float v8f __attribute__((ext_vector_type(8)));

// D = A(16x4 f32) * B(4x16 f32) + C(16x16 f32), wave32 WMMA
__device__ __forceinline__ v8f wmma_f32x4(v2f a, v2f b, v8f c) {
  return __builtin_amdgcn_wmma_f32_16x16x4_f32(
      /*neg_a=*/false, a, /*neg_b=*/false, b,
      /*c_mod=*/(short)0, c, /*reuse_a=*/false, /*reuse_b=*/false);
}

__device__ __forceinline__ int imin(int a, int b) { return a < b ? a : b; }

// ---------------------------------------------------------------------------
// Kernel 1 & 4: C[M,N] = A[M,K] @ W[N,K]^T   (nn.Linear, bias=False)
// grid: (ceil(M/16), N/128), block: 256 threads = 8 waves, 1 wave -> 16x16 tile
// OOB A rows are CLAMPED (not zeroed): garbage only reaches C rows >= M,
// which are guarded at the store. Keeps the hot loop free of exec churn.
// ---------------------------------------------------------------------------
__global__ __launch_bounds__(256)
void gemm_nt(const float* __restrict__ A, const float* __restrict__ W,
             float* __restrict__ C, int M, int N, int K) {
  const int lane = threadIdx.x & 31;
  const int wave = threadIdx.x >> 5;
  const int l    = lane & 15;   // row (A) / col (B) within tile
  const int half = lane >> 4;   // selects K pair {0,1} vs {2,3}
  const int m0 = blockIdx.x * 16;
  const int n0 = blockIdx.y * 128 + wave * 16;
  if (n0 >= N) return;  // wave-uniform

  const int arow = imin(m0 + l, M - 1);          // clamped, always in-bounds
  const float* ap = A + (size_t)arow * K + 2 * half;
  const float* wp = W + (size_t)(n0 + l) * K + 2 * half;

  __builtin_prefetch(ap, 0, 1);   // global_prefetch_b8
  __builtin_prefetch(wp, 0, 1);

  v8f acc = {};
  #pragma unroll 8
  for (int k = 0; k < K; k += 4) {
    const v2f a = *(const v2f*)(ap + k);
    const v2f b = *(const v2f*)(wp + k);
    acc = wmma_f32x4(a, b, acc);
  }

  #pragma unroll
  for (int r = 0; r < 8; ++r) {
    const int row = m0 + r + 8 * half;   // C layout: VGPR r -> M=r / M=8+r
    if (row < M) C[(size_t)row * N + (n0 + l)] = acc[r];
  }
}

// ---------------------------------------------------------------------------
// Kernel 2: in-place interleaved RoPE on q (cols 0..511) and k (cols 512..1023)
// of proj[B*S, 1536]. One thread per (b,s,qk,h,pair).
// ---------------------------------------------------------------------------
__global__ void rope_inplace(float* __restrict__ proj) {
  const int idx = blockIdx.x * blockDim.x + threadIdx.x;
  const int total = M_ROWS * 2 * 256;  // q and k halves, 256 pairs each
  if (idx >= total) return;
  const int i  = idx & 31;          // pair index within head
  const int h  = (idx >> 5) & 7;    // head
  const int qk = (idx >> 8) & 1;    // 0 = q, 1 = k
  const int bs = idx >> 9;          // flattened (b*S + s)
  const int s  = bs % S_LEN;

  // inv_freq = 10000^(-i/32);  ln(10000) = 9.210340371976184
  const float inv = __expf(-(float)i * (9.210340371976184f / 32.0f));
  const float ang = (float)s * inv;
  float sn, cs;
  sincosf(ang, &sn, &cs);

  const size_t base = (size_t)bs * PROJ_N + qk * HIDDEN + h * HEAD_DIM + 2 * i;
  const float x0 = proj[base];
  const float x1 = proj[base + 1];
  proj[base]     = x0 * cs - x1 * sn;
  proj[base + 1] = x1 * cs + x0 * sn;
}

// ---------------------------------------------------------------------------
// Kernel 3: masked-softmax attention for one (b, h, 16-query tile).
// grid: (65, H, B), block: 128 threads = 4 waves.
// proj layout per row: [q(512) | k(512) | v(512)], q/k already RoPE'd.
// ctx layout: [B, S, H*hd] (ready for out-projection GEMM).
// All OOB row indices are clamped; invalid results are overwritten by the
// branchless mask select (scores) or carry exactly-zero attn weight (V).
// ---------------------------------------------------------------------------
__global__ __launch_bounds__(128)
void attn_kernel(const float* __restrict__ proj,
                 const unsigned char* __restrict__ mask,
                 float* __restrict__ ctx) {
  __shared__ float sc[16 * S_PAD];   // score strip, then exp(scores)
  __shared__ float part[16 * 8];
  __shared__ float srowmax[16];
  __shared__ float srowsum[16];

  const int qt = blockIdx.x;
  const int h  = blockIdx.y;
  const int b  = blockIdx.z;
  const int tid  = threadIdx.x;
  const int lane = tid & 31;
  const int wave = tid >> 5;
  const int l    = lane & 15;
  const int half = lane >> 4;
  const int q0   = qt * 16;

  const float* base = proj + (size_t)b * S_LEN * PROJ_N;
  const int qcol = h * HEAD_DIM;
  const int kcol = HIDDEN + h * HEAD_DIM;
  const int vcol = 2 * HIDDEN + h * HEAD_DIM;

  // ---- Phase 1: scores = (Q @ K^T) * scale, masked, into LDS ----
  const int qrow = imin(q0 + l, S_LEN - 1);      // clamped
  const float* qp = base + (size_t)qrow * PROJ_N + qcol + 2 * half;

  for (int kt = wave; kt < S_TILES; kt += 4) {
    const int k0   = kt * 16;
    const int krow = imin(k0 + l, S_LEN - 1);    // clamped
    const float* kp = base + (size_t)krow * PROJ_N + kcol + 2 * half;

    v8f acc = {};
    #pragma unroll
    for (int kd = 0; kd < HEAD_DIM; kd += 4) {
      const v2f a  = *(const v2f*)(qp + kd);
      const v2f bb = *(const v2f*)(kp + kd);
      acc = wmma_f32x4(a, bb, acc);
    }
    #pragma unroll
    for (int r = 0; r < 8; ++r) {
      const int lr  = r + 8 * half;
      const int q   = q0 + lr;
      const int key = k0 + l;
      // clamped mask fetch, branchless validity select
      const size_t midx = (size_t)imin(q, S_LEN - 1) * S_LEN + imin(key, S_LEN - 1);
      const bool valid = (q < S_LEN) && (key < S_LEN) && (mask[midx] != 0);
      sc[lr * S_PAD + key] = valid ? acc[r] * 0.125f : -1e30f;
    }
  }
  __syncthreads();

  // ---- Phase 2: row-wise softmax over S_PAD columns (8 threads / row) ----
  const int row  = tid >> 3;
  const int slot = tid & 7;
  float m = -1e30f;
  for (int c = slot; c < S_PAD; c += 8) m = fmaxf(m, sc[row * S_PAD + c]);
  part[row * 8 + slot] = m;
  __syncthreads();
  if (tid < 16) {
    float mm = part[tid * 8];
    #pragma unroll
    for (int j = 1; j < 8; ++j) mm = fmaxf(mm, part[tid * 8 + j]);
    srowmax[tid] = mm;
  }
  __syncthreads();
  const float rmax = srowmax[row];
  float sum = 0.f;
  for (int c = slot; c < S_PAD; c += 8) {
    const float e = __expf(sc[row * S_PAD + c] - rmax);  // masked -> exactly 0
    sc[row * S_PAD + c] = e;
    sum += e;
  }
  part[row * 8 + slot] = sum;
  __syncthreads();
  if (tid < 16) {
    float ss = part[tid * 8];
    #pragma unroll
    for (int j = 1; j < 8; ++j) ss += part[tid * 8 + j];
    srowsum[tid] = ss;
  }
  __syncthreads();

  // ---- Phase 3: out = softmax @ V, one 16-wide hd slice per wave ----
  const int n0 = wave * 16;  // 4 waves cover hd = 64
  v8f acc = {};
  #pragma unroll 4
  for (int kk = 0; kk < S_PAD; kk += 4) {
    // A fragment (attn) from LDS: row = l, K pair = kk + 2*half
    const v2f a = *(const v2f*)(&sc[l * S_PAD + kk + 2 * half]);
    // B fragment (V): rows kk+2*half, +1 (clamped; attn weight is 0 there)
    const int key0 = imin(kk + 2 * half, S_LEN - 1);
    const int key1 = imin(kk + 2 * half + 1, S_LEN - 1);
    v2f bb;
    bb.x = base[(size_t)key0 * PROJ_N + vcol + n0 + l];
    bb.y = base[(size_t)key1 * PROJ_N + vcol + n0 + l];
    acc = wmma_f32x4(a, bb, acc);
  }
  #pragma unroll
  for (int r = 0; r < 8; ++r) {
    const int lr = r + 8 * half;
    const int q  = q0 + lr;
    if (q < S_LEN) {
      const float invsum = 1.0f / srowsum[lr];
      ctx[((size_t)(b * S_LEN + q)) * HIDDEN + h * HEAD_DIM + n0 + l] =
          acc[r] * invsum;
    }
  }
}

// ---------------------------------------------------------------------------
// Launch: qkv-proj GEMM -> RoPE -> attention -> out-proj GEMM
// ---------------------------------------------------------------------------
extern "C" void kernel_launch(void* const* d_in, const int* in_sizes, int n_in,
                              void* d_out, int out_size, void* d_ws, size_t ws_size,
                              hipStream_t stream) {
  const float* x     = (const float*)d_in[0];          // [B, S, 512]
  const float* w_in  = (const float*)d_in[1];          // [1536, 512]
  const float* w_out = (const float*)d_in[2];          // [512, 512]
  const unsigned char* mask = (const unsigned char*)d_in[3];  // [S, S] bool
  float* out = (float*)d_out;                          // [B, S, 512]

  float* proj = (float*)d_ws;                          // [8200, 1536]
  float* ctx  = proj + (size_t)M_ROWS * PROJ_N;        // [8200, 512]

  // 1) proj = x @ w_in^T
  gemm_nt<<<dim3((M_ROWS + 15) / 16, PROJ_N / 128), 256, 0, stream>>>(
      x, w_in, proj, M_ROWS, PROJ_N, HIDDEN);

  // 2) RoPE in place on q,k halves of proj
  {
    const int total = M_ROWS * 2 * 256;
    rope_inplace<<<(total + 255) / 256, 256, 0, stream>>>(proj);
  }

  // 3) attention -> ctx [B, S, 512]
  attn_kernel<<<dim3(S_TILES, N_HEAD, BATCH), 128, 0, stream>>>(proj, mask, ctx);

  // 4) out = ctx @ w_out^T
  gemm_nt<<<dim3((M_ROWS + 15) / 16, HIDDEN / 128), 256, 0, stream>>>(
      ctx, w_out, out, M_ROWS, HIDDEN, HIDDEN);
}